// Rule_weight_layer_25039659335977
// MI455X (gfx1250) — compile-verified
//
#include <hip/hip_runtime.h>

// ---------------------------------------------------------------------------
// Rule-weight layer for MI455X (gfx1250, wave32).
// w_k = theta_k * 2^( sum_i delta[k,i] * log2(alpha[k,i]) ); out = w / sum(w)
// Memory-bound: ~272 MB total traffic -> ~11.7 us floor @ 23.3 TB/s.
//  - streaming inputs: global_load_b128 with NT hint (268 MB > 192 MB L2)
//  - inner loop: v_log_f32 (TRANS) co-executing with v_fmac_f32
//  - wave sums: V_WMMA_F32_16X16X4_F32 ones-matrix reduction (exact fp32),
//    finished with 4 pure-xor ds_swizzle steps (lanes 0-15 of D VGPR0 hold a
//    complete copy of the 16 column sums; rows 0 and 8 are identical).
// ---------------------------------------------------------------------------

typedef __attribute__((ext_vector_type(2))) float v2f;
typedef __attribute__((ext_vector_type(4))) float f4;
typedef __attribute__((ext_vector_type(8))) float v8f;

#define NUM_K 524288
#define NUM_A 64
#define B1 256
#define NBLK (NUM_K / B1) /* 2048 */

// ds_swizzle group-of-32 pure-xor pattern: new_lane = lane ^ xor_mask
// offset = (xor_mask << 10) | (or_mask << 5) | and_mask, and_mask = 0x1f.
#define SWZ_XOR(x) (((x) << 10) | 0x1f)

template <int IMM>
__device__ __forceinline__ float xor_add(float s) {
  // ds_swizzle_b32: one DS op, no cmp/cndmask; all lanes active here.
  // IMM is a template constant -> satisfies the builtin's immediate operand.
  return s + __int_as_float(
                 __builtin_amdgcn_ds_swizzle(__float_as_int(s), IMM));
}

// Exact fp32 sum of one value per lane across the wave32. Requires EXEC
// all-1s (call only from fully-populated, non-divergent code).
// A = ones(16x4), B holds the 32 lane values (B-VGPR0 = v, B-VGPR1 = 0):
// D[m,n] = sum_k B[k,n], all D rows identical. d[0] lanes 0-15 = the 16
// column sums (lanes 16-31 = duplicate copy). Summing within each 16-lane
// half (xor 8,4,2,1 never crosses halves) puts the wave total in every lane.
__device__ __forceinline__ float wave_sum32(float v) {
  v2f a;
  a[0] = 1.0f;
  a[1] = 1.0f;
  v2f b;
  b[0] = v;
  b[1] = 0.0f;
  v8f c = {};
  v8f d = __builtin_amdgcn_wmma_f32_16x16x4_f32(
      /*neg_a=*/false, a, /*neg_b=*/false, b,
      /*c_mod=*/(short)0, c, /*reuse_a=*/false, /*reuse_b=*/false);
  float s = d[0];
  s = xor_add<SWZ_XOR(8)>(s);
  s = xor_add<SWZ_XOR(4)>(s);
  s = xor_add<SWZ_XOR(2)>(s);
  s = xor_add<SWZ_XOR(1)>(s);
  return s;  // wave total, uniform across all 32 lanes
}

__global__ __launch_bounds__(B1) void rw_compute(
    const float* __restrict__ alpha, const float* __restrict__ theta,
    const float* __restrict__ delta, float* __restrict__ w,
    float* __restrict__ partials) {
  const int k = blockIdx.x * B1 + threadIdx.x;  // grid exactly covers NUM_K

  const f4* a4 = (const f4*)(alpha + (size_t)k * NUM_A);
  const f4* d4 = (const f4*)(delta + (size_t)k * NUM_A);

  float s = 0.0f;
#pragma unroll
  for (int i = 0; i < NUM_A / 4; ++i) {
    const f4 av = __builtin_nontemporal_load(a4 + i);  // global_load_b128 NT
    const f4 dv = __builtin_nontemporal_load(d4 + i);
    s = fmaf(dv[0], __builtin_amdgcn_logf(av[0]), s);  // v_log_f32 = log2
    s = fmaf(dv[1], __builtin_amdgcn_logf(av[1]), s);
    s = fmaf(dv[2], __builtin_amdgcn_logf(av[2]), s);
    s = fmaf(dv[3], __builtin_amdgcn_logf(av[3]), s);
  }
  const float wk = theta[k] * __builtin_amdgcn_exp2f(s);  // v_exp_f32 = exp2
  w[k] = wk;  // RT hint: stays in L2 for the normalize pass

  // Block sum: WMMA wave reduction (EXEC all-1s here), then LDS across 8 waves.
  const float wsum = wave_sum32(wk);
  __shared__ float sm[B1 / 32];
  const int lane = threadIdx.x & 31;
  const int wv = threadIdx.x >> 5;
  if (lane == 0) sm[wv] = wsum;
  __syncthreads();
  if (threadIdx.x == 0) {
    float t = 0.0f;
#pragma unroll
    for (int i = 0; i < B1 / 32; ++i) t += sm[i];
    partials[blockIdx.x] = t;
  }
}

__global__ __launch_bounds__(256) void rw_reduce(
    const float* __restrict__ partials, float* __restrict__ inv_out) {
  float s = 0.0f;
#pragma unroll
  for (int i = 0; i < NBLK / 256; ++i)
    s += partials[threadIdx.x + i * 256];  // coalesced, fixed order
  const float wsum = wave_sum32(s);        // EXEC all-1s: full 256-thread block
  __shared__ float sm[256 / 32];
  const int lane = threadIdx.x & 31;
  const int wv = threadIdx.x >> 5;
  if (lane == 0) sm[wv] = wsum;
  __syncthreads();
  if (threadIdx.x == 0) {
    float t = 0.0f;
#pragma unroll
    for (int i = 0; i < 256 / 32; ++i) t += sm[i];
    inv_out[0] = 1.0f / t;
  }
}

__global__ __launch_bounds__(256) void rw_scale(float* __restrict__ w,
                                                const float* __restrict__ inv_p) {
  const float inv = inv_p[0];  // uniform -> scalar load
  const int idx = blockIdx.x * 256 + threadIdx.x;
  f4* w4 = (f4*)w;
  f4 v = w4[idx];
  v[0] *= inv;
  v[1] *= inv;
  v[2] *= inv;
  v[3] *= inv;
  w4[idx] = v;
}

extern "C" void kernel_launch(void* const* d_in, const int* in_sizes, int n_in,
                              void* d_out, int out_size, void* d_ws,
                              size_t ws_size, hipStream_t stream) {
  (void)in_sizes;
  (void)n_in;
  (void)out_size;
  (void)ws_size;
  const float* alpha = (const float*)d_in[0];
  const float* theta = (const float*)d_in[1];
  const float* delta = (const float*)d_in[2];
  float* out = (float*)d_out;

  float* partials = (float*)d_ws;  // NBLK floats (written before read)
  float* inv = partials + NBLK;    // 1 float

  rw_compute<<<NBLK, B1, 0, stream>>>(alpha, theta, delta, out, partials);
  rw_reduce<<<1, 256, 0, stream>>>(partials, inv);
  rw_scale<<<NUM_K / 4 / 256, 256, 0, stream>>>(out, inv);
}